// CCN_2D_31404800868658
// MI455X (gfx1250) — compile-verified
//
#include <hip/hip_runtime.h>

// CCN-2D on k-regular graph, N=256, K=16, C0=16, H=32.
// Strategy: per-vertex blocks; all 16x16 contractions + the feature GEMM use
// V_WMMA_F32_16X16X4_F32 (full f32 precision, CDNA5 wave32 WMMA).

typedef float v2f __attribute__((ext_vector_type(2)));
typedef float v8f __attribute__((ext_vector_type(8)));

#define WMMA4(a, b, c) \
  __builtin_amdgcn_wmma_f32_16x16x4_f32(false, (a), false, (b), (short)0, (c), false, false)

// ---------------------------------------------------------------------------
// Effective weights: contract18 with A=I collapses 18 blocks -> 8 features.
//   f0: sum_b        coeff 16*W[0] + W[5] + 16*(W[6]+...+W[14])
//   f1: sum_ab (bcast j)      W[1]
//   f2: sum_t T          16 * W[2]
//   f3: sum_tb (bcast j)      W[3]
//   f4: delta_ij * sum_all    W[4]
//   f5: T[i,j,j]              W[15]
//   f6: T[j,i,j]              W[16]
//   f7: delta_ij * sum T[t,t,t]  W[17]
// ---------------------------------------------------------------------------
__device__ __forceinline__ float weffval(const float* W, int C, int f, int c, int h) {
  const int H = 32;
#define WROW(m) W[((m) * C + c) * H + h]
  switch (f) {
    case 0: {
      float s = 16.0f * WROW(0) + WROW(5);
      for (int m = 6; m < 15; ++m) s += 16.0f * WROW(m);
      return s;
    }
    case 1: return WROW(1);
    case 2: return 16.0f * WROW(2);
    case 3: return WROW(3);
    case 4: return WROW(4);
    case 5: return WROW(15);
    case 6: return WROW(16);
    default: return WROW(17);
  }
#undef WROW
}

__global__ void k_prep(const float* __restrict__ W1, const float* __restrict__ W2,
                       float* __restrict__ Weff1, float* __restrict__ Weff2) {
  int idx = blockIdx.x * blockDim.x + threadIdx.x;
  int stride = gridDim.x * blockDim.x;
  for (int e = idx; e < 8 * 16 * 32; e += stride) {
    int h = e & 31, row = e >> 5, f = row >> 4, c = row & 15;
    Weff1[e] = weffval(W1, 16, f, c, h);
  }
  for (int e = idx; e < 8 * 32 * 32; e += stride) {
    int h = e & 31, row = e >> 5, f = row >> 5, c = row & 31;
    Weff2[e] = weffval(W2, 32, f, c, h);
  }
}

// F0[n,p,q,c] = X[n,c]
__global__ void k_bcast(const float* __restrict__ X, float* __restrict__ F0) {
  int idx = blockIdx.x * blockDim.x + threadIdx.x;
  int stride = gridDim.x * blockDim.x;
  for (int e = idx; e < 256 * 16 * 16 * 16; e += stride) {
    int c = e & 15;
    int n = e >> 12;
    F0[e] = X[n * 16 + c];
  }
}

// ---------------------------------------------------------------------------
// f is a compile-time constant at every call site (fully unrolled K loop), so
// the switch folds to a single LDS load / select.
__device__ __forceinline__ float featval(int f, int i, int j, int c,
                                         const float* SB, const float* SAB,
                                         const float* STm, const float* STB,
                                         const float* SALL, const float* DAB,
                                         const float* D2S, const float* DTTT) {
  switch (f) {
    case 0: return SB[(i * 16 + j) * 16 + c];
    case 1: return SAB[i * 16 + c];
    case 2: return STm[(i * 16 + j) * 16 + c];
    case 3: return STB[i * 16 + c];
    case 4: return (i == j) ? SALL[c] : 0.0f;
    case 5: return DAB[(i * 16 + j) * 16 + c];
    case 6: return D2S[(j * 16 + i) * 16 + c];
    default: return (i == j) ? DTTT[c] : 0.0f;
  }
}

// One block per vertex n; blockIdx.y selects a 16-channel slice of the input.
// Writes pre-activation partials: Pre[cb][n][i][j][h] (plain stores, no atomics).
template <int CT>
__global__ __launch_bounds__(128) void k_layer(const float* __restrict__ Fin,  // (256,16,16,CT)
                                               const int* __restrict__ nbrs,   // (256,16)
                                               const float* __restrict__ Weff, // (8*CT,32)
                                               float* __restrict__ Pre) {      // (CT/16,256,16,16,32)
  constexpr int H = 32;
  const int n = blockIdx.x;
  const int cb = blockIdx.y;
  const int c0g = cb * 16;
  const int tid = threadIdx.x;
  const int wave = tid >> 5;
  const int lane = tid & 31;
  const int l16 = lane & 15;
  const int hi = lane >> 4;

  __shared__ int nbr[16];
  __shared__ int nn[256];         // NN[t][p] = nbrs[nbr[t]][p]
  __shared__ float chS[4096];     // CH[t][a][p] in {0,1}
  __shared__ float Fst[4096];     // F_j[p][q][cl], 16 local channels
  __shared__ float stage[4][256]; // per-wave 16x16 bounce tile
  __shared__ float SB[4096];      // sum_b  [t][a][cl]
  __shared__ float STm[4096];     // sum_t T [a][b][cl]
  __shared__ float DAB[4096];     // T[t][e][e]  -> [t][e][cl]
  __shared__ float D2S[4096];     // T[t][a][t]  -> [t][a][cl]
  __shared__ float SAB[256];      // sum_ab [t][cl]  (derived post-loop)
  __shared__ float STB[256];      // sum_tb [a][cl]  (derived post-loop)
  __shared__ float SALL[16];
  __shared__ float DTTT[16];
  __shared__ float WeffS[128 * 32]; // local rows kl = f*16+cl

  if (tid < 16) nbr[tid] = nbrs[n * 16 + tid];
  __syncthreads();
  for (int e = tid; e < 256; e += 128) nn[e] = nbrs[nbr[e >> 4] * 16 + (e & 15)];
  for (int e = tid; e < 128 * 32; e += 128) {
    int kl = e >> 5, h = e & 31, f = kl >> 4, cl = kl & 15;
    WeffS[e] = Weff[(f * CT + c0g + cl) * H + h];
  }
  __syncthreads();
  for (int e = tid; e < 4096; e += 128) {
    int t = e >> 8, a = (e >> 4) & 15, p = e & 15;
    chS[e] = (nbr[a] == nn[t * 16 + p]) ? 1.0f : 0.0f;
  }

  v8f accT[4];
#pragma unroll
  for (int cl = 0; cl < 4; ++cl) accT[cl] = (v8f){0, 0, 0, 0, 0, 0, 0, 0};

  for (int t = 0; t < 16; ++t) {
    __syncthreads(); // protect Fst reuse (and chS on first iter)
    const float* src = Fin + (size_t)nbr[t] * (256 * CT) + c0g;
    for (int e = tid; e < 1024; e += 128) { // 4096 floats as float4
      int pq = e >> 2;
      int c4 = (e & 3) * 4;
      float4 v = *reinterpret_cast<const float4*>(src + pq * CT + c4);
      *reinterpret_cast<float4*>(&Fst[pq * 16 + c4]) = v;
    }
    __syncthreads();

    // CH_t chunks in WMMA A layout (lane=a, vgpr pair = p0/p0+1); same values
    // serve as the B operand of matmul-2 (lane=b, K rows q0/q0+1).
    v2f chA[4];
#pragma unroll
    for (int kk = 0; kk < 4; ++kk) {
      int p0 = kk * 4 + hi * 2;
      chA[kk].x = chS[t * 256 + l16 * 16 + p0];
      chA[kk].y = chS[t * 256 + l16 * 16 + p0 + 1];
    }
    float* st = stage[wave];

#pragma unroll
    for (int cl = 0; cl < 4; ++cl) {
      int c = wave + 4 * cl; // local channel
      // tmp = CH_t @ F_j[:,:,c]
      v8f t1 = {0, 0, 0, 0, 0, 0, 0, 0};
#pragma unroll
      for (int kk = 0; kk < 4; ++kk) {
        int p0 = kk * 4 + hi * 2;
        v2f b;
        b.x = Fst[(p0 * 16 + l16) * 16 + c];
        b.y = Fst[((p0 + 1) * 16 + l16) * 16 + c];
        t1 = WMMA4(chA[kk], b, t1);
      }
      // bounce tmp (C/D layout) -> [a][q] tile in LDS
#pragma unroll
      for (int v = 0; v < 8; ++v) st[(v + 8 * hi) * 16 + l16] = t1[v];
      // T = tmp @ CH_t^T
      v8f t2 = {0, 0, 0, 0, 0, 0, 0, 0};
#pragma unroll
      for (int kk = 0; kk < 4; ++kk) {
        int q0 = kk * 4 + hi * 2;
        v2f a2;
        a2.x = st[l16 * 16 + q0];
        a2.y = st[l16 * 16 + q0 + 1];
        t2 = WMMA4(a2, chA[kk], t2);
      }
      accT[cl] += t2; // sum over t stays in the accumulator
      // dump T_t tile for the reductions
#pragma unroll
      for (int v = 0; v < 8; ++v) st[(v + 8 * hi) * 16 + l16] = t2[v];
      if (hi == 0) { // lane a = l16
        float rs = 0.0f;
#pragma unroll
        for (int b = 0; b < 16; ++b) rs += st[l16 * 16 + b];
        SB[(t * 16 + l16) * 16 + c] = rs;                  // sum_b
        DAB[(t * 16 + l16) * 16 + c] = st[l16 * 16 + l16]; // T[t,a,a]
        D2S[(t * 16 + l16) * 16 + c] = st[l16 * 16 + t];   // T[t,a,t]
      }
    }
  }

  // dump per-wave accumulators (sum_t T)
#pragma unroll
  for (int cl = 0; cl < 4; ++cl) {
    int c = wave + 4 * cl;
#pragma unroll
    for (int v = 0; v < 8; ++v) STm[((v + 8 * hi) * 16 + l16) * 16 + c] = accT[cl][v];
  }
  __syncthreads();
  // SAB[t][c] = sum_a SB[t][a][c];  STB[a][c] = sum_t SB[t][a][c]
  for (int e = tid; e < 256; e += 128) {
    int x = e >> 4, c = e & 15;
    float sab = 0.0f, stb = 0.0f;
    for (int y = 0; y < 16; ++y) {
      sab += SB[(x * 16 + y) * 16 + c]; // x=t, y=a
      stb += SB[(y * 16 + x) * 16 + c]; // x=a, y=t
    }
    SAB[e] = sab;
    STB[e] = stb;
  }
  __syncthreads();
  // SALL[c] = sum_t SAB[t][c];  DTTT[c] = sum_t DAB[t][t][c]
  if (tid < 16) {
    float sall = 0.0f, dttt = 0.0f;
    for (int t = 0; t < 16; ++t) {
      sall += SAB[t * 16 + tid];
      dttt += DAB[(t * 16 + t) * 16 + tid];
    }
    SALL[tid] = sall;
    DTTT[tid] = dttt;
  }
  __syncthreads();

  // Feature GEMM: 256 rows (i,j) x 32 cols, K = 8*16 local features, via WMMA.
  float* preOut = Pre + (size_t)cb * (256 * 256 * H) + (size_t)n * (256 * H);
  for (int pair = wave; pair < 32; pair += 4) {
    int mt = pair >> 1, nt = pair & 1;
    int i = mt;
    int j = l16;
    int h = nt * 16 + l16;
    v8f acc = {0, 0, 0, 0, 0, 0, 0, 0};
#pragma unroll
    for (int kk = 0; kk < 32; ++kk) {
      int k0 = kk * 4 + hi * 2; // even; (k0+1) stays within the same feature f
      int f = (kk * 4) >> 4;    // uniform & compile-time per iteration
      int c = k0 & 15;
      v2f a2, b2;
      a2.x = featval(f, i, j, c, SB, SAB, STm, STB, SALL, DAB, D2S, DTTT);
      a2.y = featval(f, i, j, c + 1, SB, SAB, STm, STB, SALL, DAB, D2S, DTTT);
      b2.x = WeffS[k0 * 32 + h];
      b2.y = WeffS[(k0 + 1) * 32 + h];
      acc = WMMA4(a2, b2, acc);
    }
#pragma unroll
    for (int v = 0; v < 8; ++v) {
      int jr = v + 8 * hi;
      preOut[(i * 16 + jr) * H + h] = acc[v];
    }
  }
}

// bias + relu -> F1
__global__ void k_act1(const float* __restrict__ Pre, const float* __restrict__ b,
                       float* __restrict__ F1) {
  int idx = blockIdx.x * blockDim.x + threadIdx.x;
  int stride = gridDim.x * blockDim.x;
  for (int e = idx; e < 2097152; e += stride) {
    float v = Pre[e] + b[e & 31];
    F1[e] = v > 0.0f ? v : 0.0f;
  }
}

// layer2: combine the two channel-half partials, bias+relu, dot with fc_w,
// block-reduce deterministically into partials[].
__global__ __launch_bounds__(256) void k_part(const float* __restrict__ Pre,
                                              const float* __restrict__ b,
                                              const float* __restrict__ fc_w,
                                              float* __restrict__ partials) {
  __shared__ float red[256];
  int tid = threadIdx.x;
  float s = 0.0f;
  for (int e = blockIdx.x * 256 + tid; e < 2097152; e += gridDim.x * 256) {
    int h = e & 31;
    float v = Pre[e] + Pre[e + 2097152] + b[h];
    v = v > 0.0f ? v : 0.0f;
    s += v * fc_w[h];
  }
  red[tid] = s;
  __syncthreads();
  for (int off = 128; off > 0; off >>= 1) {
    if (tid < off) red[tid] += red[tid + off];
    __syncthreads();
  }
  if (tid == 0) partials[blockIdx.x] = red[0];
}

__global__ void k_final(const float* __restrict__ partials, int np,
                        const float* __restrict__ fc_b, float* __restrict__ out) {
  if (blockIdx.x == 0 && threadIdx.x == 0) {
    float s = 0.0f;
    for (int i = 0; i < np; ++i) s += partials[i];
    out[0] = s + fc_b[0];
  }
}

// ---------------------------------------------------------------------------
extern "C" void kernel_launch(void* const* d_in, const int* in_sizes, int n_in,
                              void* d_out, int out_size, void* d_ws, size_t ws_size,
                              hipStream_t stream) {
  const float* X = (const float*)d_in[0];
  const int* nbrs = (const int*)d_in[1];
  const float* W1 = (const float*)d_in[2];
  const float* b1 = (const float*)d_in[3];
  const float* W2 = (const float*)d_in[4];
  const float* b2 = (const float*)d_in[5];
  const float* fcw = (const float*)d_in[6];
  const float* fcb = (const float*)d_in[7];

  float* ws = (float*)d_ws;
  float* F0 = ws;                   // 1,048,576 f  (256*16*16*16)
  float* F1 = F0 + 1048576;         // 2,097,152 f  (256*16*16*32)
  float* Pre = F1 + 2097152;        // 2 * 2,097,152 f (two channel-half slices)
  float* Weff1 = Pre + 2 * 2097152; // 4,096 f
  float* Weff2 = Weff1 + 4096;      // 8,192 f
  float* partials = Weff2 + 8192;   // 512 f   (total ~29.4 MB of d_ws)

  k_prep<<<8, 256, 0, stream>>>(W1, W2, Weff1, Weff2);
  k_bcast<<<512, 256, 0, stream>>>(X, F0);
  k_layer<16><<<dim3(256, 1), 128, 0, stream>>>(F0, nbrs, Weff1, Pre);
  k_act1<<<1024, 256, 0, stream>>>(Pre, b1, F1);
  k_layer<32><<<dim3(256, 2), 128, 0, stream>>>(F1, nbrs, Weff2, Pre);
  k_part<<<512, 256, 0, stream>>>(Pre, b2, fcw, partials);
  k_final<<<1, 64, 0, stream>>>(partials, 512, fcb, (float*)d_out);
}